// ABL_REG_87694642250050
// MI455X (gfx1250) — compile-verified
//
#include <hip/hip_runtime.h>
#include <math.h>

// ---------------- problem constants (from reference) ----------------
constexpr int NB    = 32;            // batch pairs
constexpr int NN    = 20;            // nodes per graph
constexpr int NPAIR = 190;           // N*(N-1)/2
constexpr int ND    = 64;            // node dim D
constexpr int NDE   = 128;           // edge/message dim DE
constexpr int EF_   = 8;
constexpr int NG    = 64;            // 2*B graphs
constexpr int TNODE = NG * NN;       // 1280
constexpr int TEDGE = NG * 60;       // 3840
constexpr int KMSG  = 2 * ND + EF_;  // 136
constexpr int KUPD  = ND + NDE;      // 192
constexpr int KLRL  = 132;           // 129 padded to multiple of 4
constexpr float INV_T = 10.0f;       // 1 / T_SINK
constexpr int SINKI = 20;

typedef __attribute__((ext_vector_type(2))) float v2f;
typedef __attribute__((ext_vector_type(8))) float v8f;

// ---------------- generic f32 WMMA GEMM: C = act(A[MxK] @ B[KxN] + bias) ----
// One wave32 per 16x16 output tile, K stepped by 4 via V_WMMA_F32_16X16X4_F32.
// Out-of-range lanes CLAMP their row/col index (A row m only feeds C row m,
// B col n only feeds C col n), so the inner loop is branch-free with EXEC
// all-ones and the A pair is a single b64 load.
__global__ __launch_bounds__(256) void gemm_wmma_f32(
    const float* __restrict__ A, const float* __restrict__ Bm,
    const float* __restrict__ bias, float* __restrict__ C,
    int M, int Nn, int K, int relu) {
  int tilesN = (Nn + 15) >> 4;
  int wave   = (blockIdx.x * blockDim.x + threadIdx.x) >> 5;
  int lane   = threadIdx.x & 31;
  int tm = wave / tilesN;
  int tn = wave - tm * tilesN;
  if (tm * 16 >= M) return;                 // wave-uniform exit

  int l15      = lane & 15;
  int mrow_raw = tm * 16 + l15;
  int ncol_raw = tn * 16 + l15;
  int mrow = mrow_raw < M  ? mrow_raw : M - 1;   // clamp: garbage rows unstored
  int ncol = ncol_raw < Nn ? ncol_raw : Nn - 1;  // clamp: garbage cols unstored
  int kg   = (lane >> 4) << 1;              // lane half selects K pair

  const float* Arow = A + (size_t)mrow * K + kg;     // 8B aligned (K even)
  const float* Bcol = Bm + (size_t)kg * Nn + ncol;

  v8f acc = {};
#pragma unroll 4
  for (int k0 = 0; k0 < K; k0 += 4) {
    v2f a = *(const v2f*)(Arow + k0);                // global_load_b64
    v2f b;
    b.x = Bcol[(size_t)k0 * Nn];
    b.y = Bcol[(size_t)(k0 + 1) * Nn];
    acc = __builtin_amdgcn_wmma_f32_16x16x4_f32(
        /*neg_a=*/false, a, /*neg_b=*/false, b,
        /*c_mod=*/(short)0, acc, /*reuse_a=*/false, /*reuse_b=*/false);
  }
  if (ncol_raw >= Nn) return;
  float bb = bias[ncol_raw];
  int rbase = tm * 16 + ((lane >> 4) << 3); // lanes 16-31 hold M = r+8
#pragma unroll
  for (int r = 0; r < 8; ++r) {
    int row = rbase + r;
    if (row < M) {
      float v = acc[r] + bb;
      if (relu) v = fmaxf(v, 0.0f);
      C[(size_t)row * Nn + ncol_raw] = v;
    }
  }
}

// ---------------- message-passing helpers ----------------
__global__ void prep_msg_input_kernel(const float* __restrict__ h,
                                      const float* __restrict__ e,
                                      const int* __restrict__ from_idx,
                                      const int* __restrict__ to_idx,
                                      float* __restrict__ m_in) {
  int idx = blockIdx.x * blockDim.x + threadIdx.x;
  if (idx >= TEDGE * KMSG) return;
  int r = idx / KMSG, c = idx - r * KMSG;
  float v;
  if (c < ND)            v = h[from_idx[r] * ND + c];
  else if (c < 2 * ND)   v = h[to_idx[r] * ND + (c - ND)];
  else                   v = e[r * EF_ + (c - 2 * ND)];
  m_in[idx] = v;
}

__global__ void seg_sum_kernel(const float* __restrict__ msg,
                               const int* __restrict__ to_idx,
                               float* __restrict__ agg) {
  int idx = blockIdx.x * blockDim.x + threadIdx.x;
  if (idx >= TEDGE * NDE) return;
  int r = idx / NDE, c = idx - r * NDE;
  atomicAdd(&agg[to_idx[r] * NDE + c], msg[idx]);
}

__global__ void prep_upd_input_kernel(const float* __restrict__ h,
                                      const float* __restrict__ agg,
                                      float* __restrict__ u_in) {
  int idx = blockIdx.x * blockDim.x + threadIdx.x;
  if (idx >= TNODE * KUPD) return;
  int r = idx / KUPD, c = idx - r * KUPD;
  u_in[idx] = (c < ND) ? h[r * ND + c] : agg[r * NDE + (c - ND)];
}

// ---------------- NC2 edge-embedding input gather ----------------
// Rows: ((g*190 + p)*2 + dir), cols: [s(64) | d(64) | adj | 0 0 0]
__global__ void prep_edge_input_kernel(const float* __restrict__ h,
                                       const float* __restrict__ qadj,
                                       const float* __restrict__ cadj,
                                       float* __restrict__ X) {
  long idx = (long)blockIdx.x * blockDim.x + threadIdx.x;
  const long total = (long)2 * NG * NPAIR * KLRL;
  if (idx >= total) return;
  int row = (int)(idx / KLRL), c = (int)(idx - (long)row * KLRL);
  int dir = row & 1, gp = row >> 1;
  int g = gp / NPAIR, p = gp - g * NPAIR;
  int i = 0, rem = p;                       // triu_indices(N,k=1) decode
  while (rem >= NN - 1 - i) { rem -= NN - 1 - i; ++i; }
  int j = i + 1 + rem;
  int s = dir ? j : i;                      // fwd: (s=i,d=j), bwd swapped
  int d = dir ? i : j;
  float v = 0.0f;
  if (c < ND)            v = h[(g * NN + s) * ND + c];
  else if (c < 2 * ND)   v = h[(g * NN + d) * ND + (c - ND)];
  else if (c == 2 * ND) {
    const float* adj = (g & 1) ? cadj : qadj;
    v = adj[(((g >> 1) * NN) + i) * NN + j];
  }
  X[idx] = v;
}

__global__ void combine_fb_kernel(const float* __restrict__ Y,
                                  float* __restrict__ E) {
  int idx = blockIdx.x * blockDim.x + threadIdx.x;
  if (idx >= NG * NPAIR * NDE) return;
  int gp = idx / NDE, c = idx - gp * NDE;
  E[idx] = Y[(2 * gp) * NDE + c] + Y[(2 * gp + 1) * NDE + c];
}

// ---------------- node sinkhorn + node alignment (one block per b) --------
__global__ __launch_bounds__(32) void node_sink_align_kernel(
    const float* __restrict__ tn, const float* __restrict__ h,
    float* __restrict__ node_align) {
  __shared__ float la[NN][NN];
  __shared__ float red[NN];
  int b = blockIdx.x, t = threadIdx.x;
  const float* tq = tn + (2 * b) * NN * NN;       // rows (2b*20..) x 20
  const float* tc = tn + (2 * b + 1) * NN * NN;
  if (t < NN) {
    for (int j = 0; j < NN; ++j) {
      float c = 0.0f;
      for (int d = 0; d < NN; ++d) c += fabsf(tq[t * NN + d] - tc[j * NN + d]);
      la[t][j] = -c * INV_T;
    }
  }
  __syncthreads();
  for (int it = 0; it < SINKI; ++it) {
    if (t < NN) {                                   // row logsumexp (axis j)
      float m = la[t][0];
      for (int j = 1; j < NN; ++j) m = fmaxf(m, la[t][j]);
      float s = 0.0f;
      for (int j = 0; j < NN; ++j) s += expf(la[t][j] - m);
      float l = m + logf(s);
      for (int j = 0; j < NN; ++j) la[t][j] -= l;
    }
    __syncthreads();
    if (t < NN) {                                   // col logsumexp (axis i)
      float m = la[0][t];
      for (int i = 1; i < NN; ++i) m = fmaxf(m, la[i][t]);
      float s = 0.0f;
      for (int i = 0; i < NN; ++i) s += expf(la[i][t] - m);
      float l = m + logf(s);
      for (int i = 0; i < NN; ++i) la[i][t] -= l;
    }
    __syncthreads();
  }
  float acc = 0.0f;
  if (t < NN) {
    const float* q = h + ((2 * b) * NN + t) * ND;
    for (int j = 0; j < NN; ++j) {
      const float* c = h + ((2 * b + 1) * NN + j) * ND;
      float nc = 0.0f;
      for (int d = 0; d < ND; ++d) nc += fabsf(q[d] - c[d]);
      acc += expf(la[t][j]) * nc;
    }
    red[t] = acc;
  }
  __syncthreads();
  if (t == 0) {
    float s = 0.0f;
    for (int i = 0; i < NN; ++i) s += red[i];
    node_align[b] = s;
  }
}

// ---------------- edge sinkhorn: 190x190 fully LDS-resident --------------
__global__ __launch_bounds__(256) void edge_sinkhorn_kernel(
    const float* __restrict__ te, float* __restrict__ plan) {
  __shared__ float la[NPAIR * NPAIR];               // 144.4 KB in 320KB LDS
  int b = blockIdx.x, t = threadIdx.x;
  const int NT = 256;
  const float* tq = te + (2 * b) * NPAIR * NN;
  const float* tc = te + (2 * b + 1) * NPAIR * NN;
  for (int idx = t; idx < NPAIR * NPAIR; idx += NT) {
    int i = idx / NPAIR, j = idx - i * NPAIR;
    float c = 0.0f;
    for (int d = 0; d < NN; ++d) c += fabsf(tq[i * NN + d] - tc[j * NN + d]);
    la[idx] = -c * INV_T;
  }
  __syncthreads();
  for (int it = 0; it < SINKI; ++it) {
    if (t < NPAIR) {                                 // row norm
      float m = la[t * NPAIR];
      for (int j = 1; j < NPAIR; ++j) m = fmaxf(m, la[t * NPAIR + j]);
      float s = 0.0f;
      for (int j = 0; j < NPAIR; ++j) s += expf(la[t * NPAIR + j] - m);
      float l = m + logf(s);
      for (int j = 0; j < NPAIR; ++j) la[t * NPAIR + j] -= l;
    }
    __syncthreads();
    if (t < NPAIR) {                                 // col norm
      float m = la[t];
      for (int i = 1; i < NPAIR; ++i) m = fmaxf(m, la[i * NPAIR + t]);
      float s = 0.0f;
      for (int i = 0; i < NPAIR; ++i) s += expf(la[i * NPAIR + t] - m);
      float l = m + logf(s);
      for (int i = 0; i < NPAIR; ++i) la[i * NPAIR + t] -= l;
    }
    __syncthreads();
  }
  for (int idx = t; idx < NPAIR * NPAIR; idx += NT)
    plan[(long)b * NPAIR * NPAIR + idx] = expf(la[idx]);
}

// ---------------- edge alignment: ec cached in LDS, wave per query row ----
// float4 (b128) traffic throughout; per-wave shuffle reduction.
__global__ __launch_bounds__(256) void edge_align_kernel(
    const float* __restrict__ E, const float* __restrict__ plan,
    const float* __restrict__ node_align, float* __restrict__ out) {
  __shared__ float ecs[NPAIR * NDE];                // 97.3 KB
  __shared__ float bsum;
  int b = blockIdx.x, t = threadIdx.x;
  const float* eq = E + (2 * b) * NPAIR * NDE;
  const float* ec = E + (2 * b + 1) * NPAIR * NDE;
  if (t == 0) bsum = 0.0f;
  {
    const float4* src = (const float4*)ec;
    float4* dst = (float4*)ecs;
    for (int idx = t; idx < NPAIR * NDE / 4; idx += 256) dst[idx] = src[idx];
  }
  __syncthreads();
  int wave = t >> 5, lane = t & 31;
  const float* pl = plan + (long)b * NPAIR * NPAIR;
  float acc = 0.0f;
  for (int i = wave; i < NPAIR; i += 8) {
    float4 q = ((const float4*)(eq + i * NDE))[lane];   // global_load_b128
    const float* plrow = pl + i * NPAIR;
    for (int j = 0; j < NPAIR; ++j) {
      float4 e4 = ((const float4*)(ecs + j * NDE))[lane]; // ds_load_b128
      float p = fabsf(q.x - e4.x) + fabsf(q.y - e4.y) +
                fabsf(q.z - e4.z) + fabsf(q.w - e4.w);
      acc += plrow[j] * p;
    }
  }
  for (int m = 16; m >= 1; m >>= 1) acc += __shfl_xor(acc, m, 32);
  if (lane == 0) atomicAdd(&bsum, acc);
  __syncthreads();
  if (t == 0) out[b] = bsum + node_align[b];
}

// ---------------- host side ----------------
static inline int ceil_div(int a, int b) { return (a + b - 1) / b; }

static void gemm(const float* A, const float* Bm, const float* bias, float* C,
                 int M, int Nn, int K, int relu, hipStream_t s) {
  int waves  = ceil_div(M, 16) * ceil_div(Nn, 16);
  int blocks = ceil_div(waves, 8);
  gemm_wmma_f32<<<blocks, 256, 0, s>>>(A, Bm, bias, C, M, Nn, K, relu);
}

extern "C" void kernel_launch(void* const* d_in, const int* in_sizes, int n_in,
                              void* d_out, int out_size, void* d_ws, size_t ws_size,
                              hipStream_t stream) {
  (void)in_sizes; (void)n_in; (void)out_size; (void)ws_size;
  const float* node_features = (const float*)d_in[0];
  const float* edge_features = (const float*)d_in[1];
  const int*   from_idx      = (const int*)d_in[2];
  const int*   to_idx        = (const int*)d_in[3];
  const float* query_adj     = (const float*)d_in[4];
  const float* corpus_adj    = (const float*)d_in[5];
  const float* enc_node_w = (const float*)d_in[6];
  const float* enc_node_b = (const float*)d_in[7];
  const float* enc_edge_w = (const float*)d_in[8];
  const float* enc_edge_b = (const float*)d_in[9];
  const float* msg_w1 = (const float*)d_in[10];
  const float* msg_b1 = (const float*)d_in[11];
  const float* msg_w2 = (const float*)d_in[12];
  const float* msg_b2 = (const float*)d_in[13];
  const float* upd_w1 = (const float*)d_in[14];
  const float* upd_b1 = (const float*)d_in[15];
  const float* upd_w2 = (const float*)d_in[16];
  const float* upd_b2 = (const float*)d_in[17];
  const float* node_sk_w1 = (const float*)d_in[18];
  const float* node_sk_b1 = (const float*)d_in[19];
  const float* node_sk_w2 = (const float*)d_in[20];
  const float* node_sk_b2 = (const float*)d_in[21];
  const float* edge_sk_w1 = (const float*)d_in[22];
  const float* edge_sk_b1 = (const float*)d_in[23];
  const float* edge_sk_w2 = (const float*)d_in[24];
  const float* edge_sk_b2 = (const float*)d_in[25];
  const float* lrl_w1 = (const float*)d_in[26];
  const float* lrl_b1 = (const float*)d_in[27];
  const float* lrl_w2 = (const float*)d_in[28];
  const float* lrl_b2 = (const float*)d_in[29];
  float* out = (float*)d_out;

  // workspace carve-up (256B aligned)
  size_t off = 0;
  auto alloc = [&](size_t nfloats) {
    float* p = (float*)((char*)d_ws + off);
    off += ((nfloats * sizeof(float) + 255) / 256) * 256;
    return p;
  };
  float* h      = alloc((size_t)TNODE * ND);
  float* e_enc  = alloc((size_t)TEDGE * EF_);
  float* m_in   = alloc((size_t)TEDGE * KMSG);
  float* msg1   = alloc((size_t)TEDGE * NDE);
  float* msg    = alloc((size_t)TEDGE * NDE);
  float* agg    = alloc((size_t)TNODE * NDE);
  float* u_in   = alloc((size_t)TNODE * KUPD);
  float* upd1   = alloc((size_t)TNODE * NDE);
  float* tn1    = alloc((size_t)TNODE * NN);
  float* tn     = alloc((size_t)TNODE * NN);
  float* nalign = alloc(NB);
  float* W1p    = alloc((size_t)KLRL * NDE);
  float* X      = alloc((size_t)2 * NG * NPAIR * KLRL);
  float* Y1     = alloc((size_t)2 * NG * NPAIR * NDE);
  float* Y2     = alloc((size_t)2 * NG * NPAIR * NDE);
  float* Emat   = alloc((size_t)NG * NPAIR * NDE);
  float* te1    = alloc((size_t)NG * NPAIR * NN);
  float* te     = alloc((size_t)NG * NPAIR * NN);
  float* plan   = alloc((size_t)NB * NPAIR * NPAIR);

  // 1) encoders
  gemm(node_features, enc_node_w, enc_node_b, h, TNODE, ND, 16, 0, stream);
  gemm(edge_features, enc_edge_w, enc_edge_b, e_enc, TEDGE, EF_, EF_, 0, stream);

  // 2) message passing x3
  for (int s = 0; s < 3; ++s) {
    prep_msg_input_kernel<<<ceil_div(TEDGE * KMSG, 256), 256, 0, stream>>>(
        h, e_enc, from_idx, to_idx, m_in);
    gemm(m_in, msg_w1, msg_b1, msg1, TEDGE, NDE, KMSG, 1, stream);
    gemm(msg1, msg_w2, msg_b2, msg, TEDGE, NDE, NDE, 0, stream);
    hipMemsetAsync(agg, 0, (size_t)TNODE * NDE * sizeof(float), stream);
    seg_sum_kernel<<<ceil_div(TEDGE * NDE, 256), 256, 0, stream>>>(msg, to_idx, agg);
    prep_upd_input_kernel<<<ceil_div(TNODE * KUPD, 256), 256, 0, stream>>>(h, agg, u_in);
    gemm(u_in, upd_w1, upd_b1, upd1, TNODE, NDE, KUPD, 1, stream);
    gemm(upd1, upd_w2, upd_b2, h, TNODE, ND, NDE, 0, stream);   // h updated
  }

  // 3) node sinkhorn transform (shared weights for q and c)
  gemm(h, node_sk_w1, node_sk_b1, tn1, TNODE, NN, ND, 1, stream);
  gemm(tn1, node_sk_w2, node_sk_b2, tn, TNODE, NN, NN, 0, stream);
  node_sink_align_kernel<<<NB, 32, 0, stream>>>(tn, h, nalign);

  // 4) NC2 edge embeddings: pad lrl_w1 129->132 rows with zeros
  hipMemsetAsync(W1p, 0, (size_t)KLRL * NDE * sizeof(float), stream);
  hipMemcpyAsync(W1p, lrl_w1, (size_t)129 * NDE * sizeof(float),
                 hipMemcpyDeviceToDevice, stream);
  {
    long total = (long)2 * NG * NPAIR * KLRL;
    prep_edge_input_kernel<<<(int)((total + 255) / 256), 256, 0, stream>>>(
        h, query_adj, corpus_adj, X);
  }
  gemm(X, W1p, lrl_b1, Y1, 2 * NG * NPAIR, NDE, KLRL, 1, stream);
  gemm(Y1, lrl_w2, lrl_b2, Y2, 2 * NG * NPAIR, NDE, NDE, 0, stream);
  combine_fb_kernel<<<ceil_div(NG * NPAIR * NDE, 256), 256, 0, stream>>>(Y2, Emat);

  // 5) edge sinkhorn transform + plan
  gemm(Emat, edge_sk_w1, edge_sk_b1, te1, NG * NPAIR, NN, NDE, 1, stream);
  gemm(te1, edge_sk_w2, edge_sk_b2, te, NG * NPAIR, NN, NN, 0, stream);
  edge_sinkhorn_kernel<<<NB, 256, 0, stream>>>(te, plan);

  // 6) fused edge transport cost + final output
  edge_align_kernel<<<NB, 256, 0, stream>>>(Emat, plan, nalign, out);
}